// KernelBuilder_19567871001260
// MI455X (gfx1250) — compile-verified
//
#include <hip/hip_runtime.h>

#define BATCH 4
#define NPTS  4096
#define DIM   256
#define KNN   11            // need 11 smallest (index 10 = k-th neighbor, k=10)
#define LOG2N 12            // NPTS = 1 << 12

typedef __attribute__((ext_vector_type(2))) float v2f;
typedef __attribute__((ext_vector_type(8))) float v8f;

// ---------------------------------------------------------------------------
// Kernel 1: per-row squared norms.  One 256-thread block per (b, i) row.
// ---------------------------------------------------------------------------
__global__ void __launch_bounds__(256) norms_kernel(const float* __restrict__ x,
                                                    float* __restrict__ norms) {
    __shared__ float red[256];
    const int row = blockIdx.x;                      // 0 .. BATCH*NPTS-1
    const float v = x[(size_t)row * DIM + threadIdx.x];
    red[threadIdx.x] = v * v;
    __syncthreads();
    for (int s = 128; s > 0; s >>= 1) {
        if (threadIdx.x < s) red[threadIdx.x] += red[threadIdx.x + s];
        __syncthreads();
    }
    if (threadIdx.x == 0) norms[row] = red[0];
}

// ---------------------------------------------------------------------------
// Kernel 2: distances via f32 WMMA.  One wave computes a 16x64 strip
// (4 accumulators sharing one A fragment).  x fits in L2, no LDS staging.
// ---------------------------------------------------------------------------
__global__ void __launch_bounds__(256) dist_kernel(const float* __restrict__ x,
                                                   const float* __restrict__ norms,
                                                   float* __restrict__ out) {
    const int lane = threadIdx.x;                    // 0..31
    const int wave = threadIdx.y;                    // 0..7
    const int b  = blockIdx.z;
    const int i0 = (blockIdx.x * 8 + wave) * 16;
    const int j0 = blockIdx.y * 64;

    const float* xb = x + (size_t)b * NPTS * DIM;
    const int m    = lane & 15;                      // M (A) / N (B) index
    const int koff = (lane >> 4) << 1;               // K pair: 0 or 2

    const float* aPtr  = xb + (size_t)(i0      + m) * DIM + koff;
    const float* bPtr0 = xb + (size_t)(j0 +  0 + m) * DIM + koff;
    const float* bPtr1 = xb + (size_t)(j0 + 16 + m) * DIM + koff;
    const float* bPtr2 = xb + (size_t)(j0 + 32 + m) * DIM + koff;
    const float* bPtr3 = xb + (size_t)(j0 + 48 + m) * DIM + koff;

    v8f acc0 = {}, acc1 = {}, acc2 = {}, acc3 = {};
    #pragma unroll 4
    for (int k = 0; k < DIM; k += 4) {
        v2f av = *(const v2f*)(aPtr  + k);
        v2f b0 = *(const v2f*)(bPtr0 + k);
        v2f b1 = *(const v2f*)(bPtr1 + k);
        v2f b2 = *(const v2f*)(bPtr2 + k);
        v2f b3 = *(const v2f*)(bPtr3 + k);
        acc0 = __builtin_amdgcn_wmma_f32_16x16x4_f32(false, av, false, b0, (short)0, acc0, false, false);
        acc1 = __builtin_amdgcn_wmma_f32_16x16x4_f32(false, av, false, b1, (short)0, acc1, false, false);
        acc2 = __builtin_amdgcn_wmma_f32_16x16x4_f32(false, av, false, b2, (short)0, acc2, false, false);
        acc3 = __builtin_amdgcn_wmma_f32_16x16x4_f32(false, av, false, b3, (short)0, acc3, false, false);
    }

    const float* nb = norms + (size_t)b * NPTS;
    const int halfM = (lane >> 4) << 3;              // C/D row group: 0 or 8
    float ni[8];
    #pragma unroll
    for (int v = 0; v < 8; ++v) ni[v] = nb[i0 + v + halfM];

    float* ob = out + (size_t)b * NPTS * NPTS;

    #define EMIT_TILE(T, ACC)                                                   \
    do {                                                                        \
        const int jc = j0 + 16 * (T) + m;                                       \
        const float nj = nb[jc];                                                \
        _Pragma("unroll")                                                       \
        for (int v = 0; v < 8; ++v) {                                           \
            float ds = ni[v] + nj - 2.0f * (ACC)[v];                            \
            ob[(size_t)(i0 + v + halfM) * NPTS + jc] =                          \
                sqrtf(fmaxf(ds, 1e-10f));                                       \
        }                                                                       \
    } while (0)

    EMIT_TILE(0, acc0);
    EMIT_TILE(1, acc1);
    EMIT_TILE(2, acc2);
    EMIT_TILE(3, acc3);
    #undef EMIT_TILE
}

// ---------------------------------------------------------------------------
// Kernel 3: 11th-smallest distance per row -> epsilon.  One block per row.
// float4 (B128) streaming reads; per-thread insertion-sorted top-11 lists,
// two-stage LDS merge.
// ---------------------------------------------------------------------------
__device__ __forceinline__ void insert11(float* a, float d) {
    if (d < a[KNN - 1]) {
        int p = KNN - 1;
        while (p > 0 && a[p - 1] > d) { a[p] = a[p - 1]; --p; }
        a[p] = d;
    }
}

__global__ void __launch_bounds__(256) kth_kernel(const float* __restrict__ dist,
                                                  float* __restrict__ eps) {
    __shared__ float cand[256 * KNN];
    __shared__ float cand2[32 * KNN];
    const int tid = threadIdx.x;
    const float4* drow = (const float4*)(dist + (size_t)blockIdx.x * NPTS);

    float best[KNN];
    #pragma unroll
    for (int t = 0; t < KNN; ++t) best[t] = 3.4e38f;
    #pragma unroll
    for (int j = 0; j < NPTS / 4 / 256; ++j) {       // 4 iterations
        float4 d4 = drow[j * 256 + tid];
        insert11(best, d4.x);
        insert11(best, d4.y);
        insert11(best, d4.z);
        insert11(best, d4.w);
    }

    #pragma unroll
    for (int t = 0; t < KNN; ++t) cand[tid * KNN + t] = best[t];
    __syncthreads();

    if (tid < 32) {
        float m2[KNN];
        #pragma unroll
        for (int t = 0; t < KNN; ++t) m2[t] = 3.4e38f;
        for (int s = 0; s < 8 * KNN; ++s) insert11(m2, cand[tid * (8 * KNN) + s]);
        #pragma unroll
        for (int t = 0; t < KNN; ++t) cand2[tid * KNN + t] = m2[t];
    }
    __syncthreads();

    if (tid == 0) {
        float m3[KNN];
        #pragma unroll
        for (int t = 0; t < KNN; ++t) m3[t] = 3.4e38f;
        for (int s = 0; s < 32 * KNN; ++s) insert11(m3, cand2[s]);
        eps[blockIdx.x] = fmaxf(m3[KNN - 1], 1e-6f);   // clip(sorted[k], 1e-6)
    }
}

// ---------------------------------------------------------------------------
// Kernel 4: Gaussian kernel values (overwrite distances in place, B128 I/O)
// + row sum q; store q^alpha = sqrt(q).  One block per row.
// ---------------------------------------------------------------------------
__global__ void __launch_bounds__(256) gauss_q_kernel(float* __restrict__ kd,
                                                      const float* __restrict__ eps,
                                                      float* __restrict__ qa) {
    __shared__ float red[256];
    const int row = blockIdx.x;                      // b*NPTS + i
    const int b   = row >> LOG2N;
    const float4* epsb4 = (const float4*)(eps + (size_t)b * NPTS);
    const float ei2 = 2.0f * eps[row];
    float4* krow = (float4*)(kd + (size_t)row * NPTS);

    float sum = 0.0f;
    #pragma unroll
    for (int j = 0; j < NPTS / 4 / 256; ++j) {       // 4 iterations
        const int idx = j * 256 + threadIdx.x;
        float4 d4 = krow[idx];
        float4 e4 = epsb4[idx];
        float4 w4;
        w4.x = expf(-(d4.x * d4.x) / (ei2 * e4.x + 1e-10f));
        w4.y = expf(-(d4.y * d4.y) / (ei2 * e4.y + 1e-10f));
        w4.z = expf(-(d4.z * d4.z) / (ei2 * e4.z + 1e-10f));
        w4.w = expf(-(d4.w * d4.w) / (ei2 * e4.w + 1e-10f));
        krow[idx] = w4;
        sum += (w4.x + w4.y) + (w4.z + w4.w);
    }
    red[threadIdx.x] = sum;
    __syncthreads();
    for (int s = 128; s > 0; s >>= 1) {
        if (threadIdx.x < s) red[threadIdx.x] += red[threadIdx.x + s];
        __syncthreads();
    }
    if (threadIdx.x == 0) qa[row] = sqrtf(red[0]);   // ALPHA = 0.5
}

// ---------------------------------------------------------------------------
// Kernel 5: two-sided degree normalization, 4 elements per thread (B128 I/O).
// ---------------------------------------------------------------------------
__global__ void __launch_bounds__(256) normalize_kernel(float* __restrict__ kd,
                                                        const float* __restrict__ qa) {
    const size_t v    = (size_t)blockIdx.x * 256 + threadIdx.x;  // float4 index
    const size_t idx  = v * 4;                                    // element index
    const int    j    = (int)(idx & (NPTS - 1));                  // j .. j+3
    const size_t rn   = idx >> LOG2N;                             // b*NPTS + i
    const size_t bbase = (rn >> LOG2N) << LOG2N;                  // b*NPTS

    const float  qi = qa[rn];
    const float4 qj = *(const float4*)(qa + bbase + (size_t)j);

    float4* p = (float4*)kd + v;
    float4 k4 = *p;
    k4.x = k4.x / (qi * qj.x + 1e-10f);
    k4.y = k4.y / (qi * qj.y + 1e-10f);
    k4.z = k4.z / (qi * qj.z + 1e-10f);
    k4.w = k4.w / (qi * qj.w + 1e-10f);
    *p = k4;
}

// ---------------------------------------------------------------------------
extern "C" void kernel_launch(void* const* d_in, const int* in_sizes, int n_in,
                              void* d_out, int out_size, void* d_ws, size_t ws_size,
                              hipStream_t stream) {
    (void)in_sizes; (void)n_in; (void)out_size; (void)ws_size;
    const float* x = (const float*)d_in[0];
    float* out = (float*)d_out;                      // (B, N, N); also distance scratch
    float* ws  = (float*)d_ws;
    float* norms = ws;                               // B*N floats
    float* eps   = ws + (size_t)BATCH * NPTS;        // B*N floats
    float* qa    = ws + (size_t)2 * BATCH * NPTS;    // B*N floats

    norms_kernel<<<BATCH * NPTS, 256, 0, stream>>>(x, norms);

    dist_kernel<<<dim3(NPTS / 128, NPTS / 64, BATCH), dim3(32, 8, 1), 0, stream>>>(
        x, norms, out);

    kth_kernel<<<BATCH * NPTS, 256, 0, stream>>>(out, eps);

    gauss_q_kernel<<<BATCH * NPTS, 256, 0, stream>>>(out, eps, qa);

    const unsigned norm_blocks = (unsigned)(((size_t)BATCH * NPTS * NPTS) / 4 / 256);
    normalize_kernel<<<norm_blocks, 256, 0, stream>>>(out, qa);
}